// Frames_Base_21947282883126
// MI455X (gfx1250) — compile-verified
//
#include <hip/hip_runtime.h>
#include <math.h>

typedef __attribute__((ext_vector_type(2))) float v2f;
typedef __attribute__((ext_vector_type(4))) float v4f;
typedef __attribute__((ext_vector_type(8))) float v8f;

#define BDIM  16
#define NDIM  8192
#define KDIM  64
#define CHUNK 256   // points staged in LDS per inner pass (16 KB of LDS)

// ---------------------------------------------------------------------------
// Phase 1: one wave32 per (batch, k-tile, N-slice).
// Computes the 16(k) x 16(moment) partial tile
//    tile[kr, m] = sum_{n in slice} W[b, n, ktile*16+kr] * P16[b, n, m]
// with V_WMMA_F32_16X16X4_F32 (f32 in, f32 acc: precision is free, we are
// memory bound). Moment rows P16 are staged per-chunk in LDS.
// ---------------------------------------------------------------------------
__global__ __launch_bounds__(32)
void frames_moments_wmma(const float* __restrict__ pnts,
                         const float* __restrict__ weights,
                         float* __restrict__ part,
                         int nslices, int per_slice) {
  __shared__ float sP[CHUNK * 16];

  const int lane = threadIdx.x;     // 0..31
  const int h    = lane >> 4;       // half-wave select (K-offset pairs)
  const int c    = lane & 15;       // row/col within half
  const int slice = blockIdx.x;
  const int kt    = blockIdx.y;     // k-tile: 0..3
  const int b     = blockIdx.z;     // batch: 0..15

  // A operand source: W[b, n, kt*16 + c], stride KDIM along n
  const float* __restrict__ Wcol =
      weights + ((size_t)b * NDIM) * KDIM + (size_t)kt * 16 + c;
  const int n0 = slice * per_slice;

  v8f acc = {0.f, 0.f, 0.f, 0.f, 0.f, 0.f, 0.f, 0.f};

  for (int cb = 0; cb < per_slice; cb += CHUNK) {
    // ---- stage CHUNK moment rows into LDS: sP[pl*16 + m] ----
    for (int t = 0; t < CHUNK / 32; ++t) {
      const int pl = lane + 32 * t;
      const float* __restrict__ pp =
          pnts + ((size_t)b * NDIM + n0 + cb + pl) * 3;
      const float x = pp[0], y = pp[1], z = pp[2];
      v4f* q = (v4f*)&sP[pl * 16];
      v4f q0 = {1.0f,  x,     y,     z    };
      v4f q1 = {x * x, x * y, x * z, y * y};
      v4f q2 = {y * z, z * z, 0.f,   0.f  };
      v4f q3 = {0.f,   0.f,   0.f,   0.f  };
      q[0] = q0; q[1] = q1; q[2] = q2; q[3] = q3;
    }
    __syncthreads();

    // ---- WMMA reduction over this chunk, 4 points per instruction ----
    const float* __restrict__ Wc = Wcol + (size_t)(n0 + cb) * KDIM;
#pragma unroll 4
    for (int i = 0; i < CHUNK / 4; ++i) {
      const int nn = 4 * i + 2 * h;
      v2f a, bb;
      // A 16x4 f32 layout: lane l -> A[l%16, 2*(l/16)+j] in VGPR j
      a[0] = Wc[(size_t)nn * KDIM];
      a[1] = Wc[(size_t)(nn + 1) * KDIM];
      // B 4x16 f32 layout: lane l -> B[2*(l/16)+j, l%16] in VGPR j
      bb[0] = sP[nn * 16 + c];
      bb[1] = sP[(nn + 1) * 16 + c];
      acc = __builtin_amdgcn_wmma_f32_16x16x4_f32(
          /*neg_a=*/false, a, /*neg_b=*/false, bb,
          /*c_mod=*/(short)0, acc, /*reuse_a=*/false, /*reuse_b=*/false);
    }
    __syncthreads();
  }

  // D layout: VGPR r, lane l -> D[r + 8*(l/16), l%16]. Store row-major tile.
  const int tile = b * 4 + kt;
  float* __restrict__ dst = part + ((size_t)tile * nslices + slice) * 256;
#pragma unroll
  for (int r = 0; r < 8; ++r)
    dst[(r + 8 * h) * 16 + c] = acc[r];
}

// ---------------------------------------------------------------------------
// Phase 2: one thread per (b,k). Fixed-order slice reduction (deterministic),
// center + covariance reconstruction, 3x3 Jacobi eigensolver, ascending sort.
// ---------------------------------------------------------------------------
__global__ __launch_bounds__(256)
void frames_finalize(const float* __restrict__ part,
                     float* __restrict__ out, int nslices) {
  const int idx = blockIdx.x * 256 + threadIdx.x;
  if (idx >= BDIM * KDIM) return;
  const int k    = idx & 63;
  const int tile = (idx >> 6) * 4 + (k >> 4);
  const int kr   = k & 15;

  float m[10];
#pragma unroll
  for (int j = 0; j < 10; ++j) m[j] = 0.f;
  for (int s = 0; s < nslices; ++s) {
    const float* p = part + ((size_t)tile * nslices + s) * 256 + kr * 16;
#pragma unroll
    for (int j = 0; j < 10; ++j) m[j] += p[j];
  }

  const float wsum = m[0];
  const float inv  = 1.0f / (wsum + 1e-6f);
  const float cx = m[1] * inv, cy = m[2] * inv, cz = m[3] * inv;

  // R = M2 - c*S1^T - S1*c^T + wsum*c*c^T  (symmetric)
  float A[3][3];
  A[0][0] = m[4] - 2.f * cx * m[1] + wsum * cx * cx;
  A[0][1] = m[5] - cx * m[2] - cy * m[1] + wsum * cx * cy;
  A[0][2] = m[6] - cx * m[3] - cz * m[1] + wsum * cx * cz;
  A[1][1] = m[7] - 2.f * cy * m[2] + wsum * cy * cy;
  A[1][2] = m[8] - cy * m[3] - cz * m[2] + wsum * cy * cz;
  A[2][2] = m[9] - 2.f * cz * m[3] + wsum * cz * cz;
  A[1][0] = A[0][1]; A[2][0] = A[0][2]; A[2][1] = A[1][2];

  float V[3][3] = {{1.f, 0.f, 0.f}, {0.f, 1.f, 0.f}, {0.f, 0.f, 1.f}};

  for (int sweep = 0; sweep < 10; ++sweep) {
    for (int pi = 0; pi < 3; ++pi) {
      for (int qi = pi + 1; qi < 3; ++qi) {
        const float apq = A[pi][qi];
        if (fabsf(apq) <= 1e-30f) continue;
        const float tau = (A[qi][qi] - A[pi][pi]) / (2.f * apq);
        const float t   = (tau >= 0.f ? 1.f : -1.f) /
                          (fabsf(tau) + sqrtf(1.f + tau * tau));
        const float cth = 1.0f / sqrtf(1.f + t * t);
        const float sth = t * cth;
#pragma unroll
        for (int r = 0; r < 3; ++r) {           // A <- A J (columns)
          const float arp = A[r][pi], arq = A[r][qi];
          A[r][pi] = cth * arp - sth * arq;
          A[r][qi] = sth * arp + cth * arq;
        }
#pragma unroll
        for (int r = 0; r < 3; ++r) {           // A <- J^T A (rows)
          const float apr = A[pi][r], aqr = A[qi][r];
          A[pi][r] = cth * apr - sth * aqr;
          A[qi][r] = sth * apr + cth * aqr;
        }
#pragma unroll
        for (int r = 0; r < 3; ++r) {           // V <- V J
          const float vrp = V[r][pi], vrq = V[r][qi];
          V[r][pi] = cth * vrp - sth * vrq;
          V[r][qi] = sth * vrp + cth * vrq;
        }
      }
    }
  }

  const float d[3] = {A[0][0], A[1][1], A[2][2]};
  int p0 = 0, p1 = 1, p2 = 2, tp;
  if (d[p0] > d[p1]) { tp = p0; p0 = p1; p1 = tp; }
  if (d[p1] > d[p2]) { tp = p1; p1 = p2; p2 = tp; }
  if (d[p0] > d[p1]) { tp = p0; p0 = p1; p1 = tp; }
  const int perm[3] = {p0, p1, p2};

  float* __restrict__ F = out;                               // [B,K,3,3]
  float* __restrict__ C = out + BDIM * KDIM * 9;             // [B,K,1,3]
  float* __restrict__ L = out + BDIM * KDIM * 9 + BDIM * KDIM * 3; // [B,K,3]
#pragma unroll
  for (int col = 0; col < 3; ++col) {
    const int j = perm[col];
    L[idx * 3 + col] = d[j];
#pragma unroll
    for (int row = 0; row < 3; ++row)
      F[idx * 9 + row * 3 + col] = V[row][j];   // eigenvectors in columns
  }
  C[idx * 3 + 0] = cx; C[idx * 3 + 1] = cy; C[idx * 3 + 2] = cz;
}

// ---------------------------------------------------------------------------
extern "C" void kernel_launch(void* const* d_in, const int* in_sizes, int n_in,
                              void* d_out, int out_size, void* d_ws,
                              size_t ws_size, hipStream_t stream) {
  (void)in_sizes; (void)n_in; (void)out_size;
  const float* pnts    = (const float*)d_in[0];   // [16, 8192, 3]  f32
  const float* weights = (const float*)d_in[1];   // [16, 8192, 64] f32
  float* part = (float*)d_ws;
  float* out  = (float*)d_out;

  // Split N into slices for occupancy; fit partials (64 tiles * nslices * 1KB)
  // into the provided workspace. Fixed-order reduction keeps determinism.
  int nslices = 32;
  while (nslices > 1 &&
         (size_t)(BDIM * 4) * nslices * 256 * sizeof(float) > ws_size)
    nslices >>= 1;
  const int per_slice = NDIM / nslices;           // multiple of CHUNK (256)

  dim3 g1(nslices, 4, BDIM);
  frames_moments_wmma<<<g1, 32, 0, stream>>>(pnts, weights, part,
                                             nslices, per_slice);
  frames_finalize<<<(BDIM * KDIM + 255) / 256, 256, 0, stream>>>(part, out,
                                                                 nslices);
}